// MultiHeadAttentionWithCache_16544214024830
// MI455X (gfx1250) — compile-verified
//
#include <hip/hip_runtime.h>
#include <math.h>

// Multi-head attention decode step with cache shift, MI455X (gfx1250).
// Memory-bound (~285 MB / ~75 MFLOP -> ~12us floor @ 23.3 TB/s):
//  - fuse cache shift-copy with the attention read (cache touched once),
//  - b64 nontemporal loads/stores for all streaming traffic,
//  - WMMA f32 16x16x4 for the projection GEMVs (full fp32 precision).

#define E 1024
#define H 16
#define D 64
#define L 16384
#define NB 256               // attention position chunks (blocks)
#define POS_PER_BLK (L / NB) // 64
#define BATCH 8

typedef __attribute__((ext_vector_type(2))) float v2f;
typedef __attribute__((ext_vector_type(8))) float v8f;

static_assert(NB == 256, "reduce kernel assumes NB==256");

// ---------------------------------------------------------------------------
// GEMV out[j0..j0+15] = x(1xE) @ W(ExE) + bias, one 16-col tile per wave,
// via V_WMMA_F32_16X16X4_F32 with A = x-chunk replicated across the 16 M rows.
// A 16x4 f32 layout: half-wave r (lane>>4) holds K=2r (vgpr0), K=2r+1 (vgpr1).
// B 4x16 layout mirrors: half-wave r holds rows K=2r / K=2r+1 at N=lane&15.
// C row M=0 lives in c[0] on lanes 0..15.
// ---------------------------------------------------------------------------
__device__ __forceinline__ void gemv16_tile(const float* __restrict__ x,
                                            const float* __restrict__ W,
                                            const float* __restrict__ bias,
                                            float* __restrict__ dst,
                                            int j0, int lane) {
  const int r   = lane >> 4;  // 0 or 1
  const int col = lane & 15;
  const float* wp = W + (size_t)(2 * r) * E + j0 + col;
  const float* xp = x + 2 * r;
  v8f c = {};
#pragma unroll 4
  for (int i0 = 0; i0 < E; i0 += 4) {
    v2f a, b;
    a.x = xp[i0];
    a.y = xp[i0 + 1];
    b.x = __builtin_nontemporal_load(wp + (size_t)i0 * E);
    b.y = __builtin_nontemporal_load(wp + (size_t)(i0 + 1) * E);
    c = __builtin_amdgcn_wmma_f32_16x16x4_f32(false, a, false, b,
                                              (short)0, c, false, false);
  }
  if (lane < 16) dst[j0 + col] = c[0] + bias[j0 + col];
}

// 24 blocks x 256 threads = 192 waves: 3 matrices x 64 column tiles.
__global__ __launch_bounds__(256) void qkv_proj_kernel(
    const float* __restrict__ x,
    const float* __restrict__ Wq, const float* __restrict__ bq,
    const float* __restrict__ Wk, const float* __restrict__ bk,
    const float* __restrict__ Wv, const float* __restrict__ bv,
    float* __restrict__ q_ws, float* __restrict__ k_ws,
    float* __restrict__ v_ws) {
  const int wave = (blockIdx.x * blockDim.x + threadIdx.x) >> 5; // 0..191
  const int lane = threadIdx.x & 31;
  const int mat  = wave >> 6;
  const int j0   = (wave & 63) << 4;
  const float *W, *b;
  float* dst;
  if (mat == 0)      { W = Wq; b = bq; dst = q_ws; }
  else if (mat == 1) { W = Wk; b = bk; dst = k_ws; }
  else               { W = Wv; b = bv; dst = v_ws; }
  gemv16_tile(x, W, b, dst, j0, lane);
}

// 8 blocks x 256 threads = 64 waves = 64 column tiles of Wo.
__global__ __launch_bounds__(256) void out_proj_kernel(
    const float* __restrict__ values,
    const float* __restrict__ Wo, const float* __restrict__ bo,
    float* __restrict__ out) {
  const int wave = (blockIdx.x * blockDim.x + threadIdx.x) >> 5; // 0..63
  const int lane = threadIdx.x & 31;
  gemv16_tile(values, Wo, bo, out, (wave & 63) << 4, lane);
}

// ---------------------------------------------------------------------------
// Fused: shift cache (read old pos+1 / new token -> write d_out), per-position
// validity mask (any v != 0 across all heads, via wave ballot + LDS OR), and
// per-head online-softmax partials over this block's position chunk.
// One wave32 per head: 32 lanes x float2 = 64 head dims. All streaming
// traffic is b64 NT per lane (256B/wave/instr); 16 waves jointly cover each
// 4KB position row, so global access stays fully coalesced.
// ---------------------------------------------------------------------------
__global__ __launch_bounds__(512) void attn_kernel(
    const float* __restrict__ cache,   // (2, L, H, D): v then k
    const float* __restrict__ q_ws,    // E
    const float* __restrict__ k_new,   // E
    const float* __restrict__ v_new,   // E
    float* __restrict__ v_out,         // L*E  (shifted v -> d_out)
    float* __restrict__ k_out,         // L*E  (shifted k -> d_out)
    float* __restrict__ part_m,        // H*NB
    float* __restrict__ part_s,        // H*NB
    float* __restrict__ part_pv) {     // H*NB*D
  __shared__ unsigned int lds_nz[H];
  const int tid  = threadIdx.x;
  const int h    = tid >> 5;
  const int lane = tid & 31;
  const int base = blockIdx.x * POS_PER_BLK;
  const int off  = h * D + 2 * lane;   // even element offset in a 1024-f row

  const v2f q2 = *(const v2f*)(q_ws + off);
  float m = -INFINITY, s = 0.f;
  v2f pv = {0.f, 0.f};

  const float* __restrict__ vsrc = cache;                  // slab 0
  const float* __restrict__ ksrc = cache + (size_t)L * E;  // slab 1

  for (int p0 = base; p0 < base + POS_PER_BLK; p0 += BATCH) {
    v2f kb[BATCH], vb[BATCH];
    unsigned int nz = 0;
#pragma unroll
    for (int j = 0; j < BATCH; ++j) {
      const int pos = p0 + j;
      v2f k2, v2;
      if (pos < L - 1) {                 // shifted read: old cache pos+1
        const size_t idx = (size_t)(pos + 1) * E + off;
        k2 = __builtin_nontemporal_load((const v2f*)(ksrc + idx));
        v2 = __builtin_nontemporal_load((const v2f*)(vsrc + idx));
      } else {                           // appended new token
        k2 = *(const v2f*)(k_new + off);
        v2 = *(const v2f*)(v_new + off);
      }
      const size_t oidx = (size_t)pos * E + off;
      __builtin_nontemporal_store(v2, (v2f*)(v_out + oidx));
      __builtin_nontemporal_store(k2, (v2f*)(k_out + oidx));
      kb[j] = k2;
      vb[j] = v2;
      if (__any((v2.x != 0.f) || (v2.y != 0.f))) nz |= (1u << j);
    }
    // combine per-head nonzero bits -> per-position mask over all 1024 elems
    if (lane == 0) lds_nz[h] = nz;
    __syncthreads();
    unsigned int combined = 0;
#pragma unroll
    for (int hh = 0; hh < H; ++hh) combined |= lds_nz[hh];
    __syncthreads();

#pragma unroll
    for (int j = 0; j < BATCH; ++j) {
      float t = q2.x * kb[j].x + q2.y * kb[j].y;
      t += __shfl_xor(t, 16, 32);        // wave32 full reduction
      t += __shfl_xor(t, 8, 32);
      t += __shfl_xor(t, 4, 32);
      t += __shfl_xor(t, 2, 32);
      t += __shfl_xor(t, 1, 32);
      if (combined & (1u << j)) {        // masked pos contributes weight 0
        const float sc = t * 0.125f;     // 1/sqrt(64)
        const float nm = fmaxf(m, sc);
        const float eo = __expf(m - nm);
        const float w  = __expf(sc - nm);
        s  = s * eo + w;
        pv = pv * eo + vb[j] * w;
        m  = nm;
      }
    }
  }
  const int pb = h * NB + blockIdx.x;
  if (lane == 0) { part_m[pb] = m; part_s[pb] = s; }
  *(v2f*)(part_pv + (size_t)pb * D + 2 * lane) = pv;
}

// One block per head: combine NB chunk partials (flash-attention merge).
__global__ __launch_bounds__(NB) void reduce_kernel(
    const float* __restrict__ part_m, const float* __restrict__ part_s,
    const float* __restrict__ part_pv, float* __restrict__ values) {
  __shared__ float red[NB];
  __shared__ float wgt[NB];
  const int h = blockIdx.x;
  const int t = threadIdx.x;

  const float m_t = part_m[h * NB + t];
  red[t] = m_t;
  __syncthreads();
  for (int sft = NB / 2; sft > 0; sft >>= 1) {
    if (t < sft) red[t] = fmaxf(red[t], red[t + sft]);
    __syncthreads();
  }
  const float M = red[0];
  __syncthreads();

  const float s_t = part_s[h * NB + t];
  const float w   = (s_t > 0.f) ? __expf(m_t - M) : 0.f;
  wgt[t] = w;
  red[t] = s_t * w;
  __syncthreads();
  for (int sft = NB / 2; sft > 0; sft >>= 1) {
    if (t < sft) red[t] += red[t + sft];
    __syncthreads();
  }
  const float S = red[0];
  __syncthreads();

  const int d = t & (D - 1);
  const int g = t >> 6;                  // 4 groups of 64 chunks
  float acc = 0.f;
  for (int b = g * 64; b < g * 64 + 64; ++b)
    acc += part_pv[(size_t)(h * NB + b) * D + d] * wgt[b];
  red[t] = acc;
  __syncthreads();
  if (t < D)
    values[h * D + t] = (red[t] + red[t + 64] + red[t + 128] + red[t + 192]) / S;
}

extern "C" void kernel_launch(void* const* d_in, const int* in_sizes, int n_in,
                              void* d_out, int out_size, void* d_ws,
                              size_t ws_size, hipStream_t stream) {
  (void)in_sizes; (void)n_in; (void)out_size; (void)ws_size;
  const float* x     = (const float*)d_in[0];
  const float* cache = (const float*)d_in[1];
  const float* Wv    = (const float*)d_in[2];
  const float* bv    = (const float*)d_in[3];
  const float* Wq    = (const float*)d_in[4];
  const float* bq    = (const float*)d_in[5];
  const float* Wk    = (const float*)d_in[6];
  const float* bk    = (const float*)d_in[7];
  const float* Wo    = (const float*)d_in[8];
  const float* bo    = (const float*)d_in[9];

  float* out   = (float*)d_out;             // [0..1023] out_i
  float* v_out = out + E;                   // shifted v cache (L*E)
  float* k_out = v_out + (size_t)L * E;     // shifted k cache (L*E)

  float* ws      = (float*)d_ws;
  float* q_ws    = ws;                      // E
  float* k_ws    = ws + E;                  // E
  float* v_ws    = ws + 2 * E;              // E
  float* values  = ws + 3 * E;              // E
  float* part_m  = ws + 4 * E;              // H*NB
  float* part_s  = part_m + H * NB;         // H*NB
  float* part_pv = part_s + H * NB;         // H*NB*D  (total ~1.1 MB)

  qkv_proj_kernel<<<24, 256, 0, stream>>>(x, Wq, bq, Wk, bk, Wv, bv,
                                          q_ws, k_ws, v_ws);
  attn_kernel<<<NB, 512, 0, stream>>>(cache, q_ws, k_ws, v_ws, v_out, k_out,
                                      part_m, part_s, part_pv);
  reduce_kernel<<<H, NB, 0, stream>>>(part_m, part_s, part_pv, values);
  out_proj_kernel<<<8, 256, 0, stream>>>(values, Wo, bo, out);
}